// SparseMemory_7430293422566
// MI455X (gfx1250) — compile-verified
//
#include <hip/hip_runtime.h>
#include <cstdint>
#include <cstddef>

// Problem constants (match reference)
#define Bsz     32
#define Mmem    65536
#define WcD     64
#define KD      8
#define IND     512
#define RD      9          // K+1
#define IFACE_D 138        // 2*Wc + R + 1
#define IFACE_P 144        // padded to 9 col-tiles of 16

typedef float v2f __attribute__((ext_vector_type(2)));
typedef float v4f __attribute__((ext_vector_type(4)));
typedef float v8f __attribute__((ext_vector_type(8)));

// ---------- top-8 (smallest) machinery; total order on (d, idx) -------------
__device__ __forceinline__ bool cless(float d0, int i0, float d1, int i1) {
  return (d0 < d1) || ((d0 == d1) && (i0 < i1));
}

// bd[0..7]/bi[0..7] kept sorted ascending by (d,idx). Fully unrolled, static
// indices only -> stays in VGPRs.
__device__ __forceinline__ void cinsert(float (&bd)[8], int (&bi)[8], float d, int i) {
  if (cless(d, i, bd[7], bi[7])) {
    bd[7] = d; bi[7] = i;
#pragma unroll
    for (int j = 7; j > 0; --j) {
      if (cless(bd[j], bi[j], bd[j - 1], bi[j - 1])) {
        float td = bd[j]; bd[j] = bd[j - 1]; bd[j - 1] = td;
        int   ti = bi[j]; bi[j] = bi[j - 1]; bi[j - 1] = ti;
      }
    }
  }
}

// ---------------------------------------------------------------------------
// Kernel 1: iface GEMM via V_WMMA_F32_16X16X4_F32 + derived small tensors.
// 18 waves: wave -> (mt in 0..1, nt in 0..8) 16x16 output tile, K-loop of 128.
// A (16x4 f32) frag per ISA 7.12.2: lane l holds {xi[m, k0+2*kp], xi[m, k0+2*kp+1]}
//   with m = l&15, kp = l>>4.  B (4x16) analogous (row K striped across lanes,
//   second half-wave offset by nrows/2 = 2).  C/D: vgpr j -> row j + 8*kp.
// Out-of-range columns (n >= 138) are handled by CLAMPING the load address
// (always in-bounds, branchless); their garbage results land only in the
// padded columns 138..143 of iface_ws, which are never read.
// ---------------------------------------------------------------------------
__global__ __launch_bounds__(576)
void sm_prep(const float* __restrict__ xi, const float* __restrict__ W,
             const float* __restrict__ bias, const float* __restrict__ read_weights,
             const float* __restrict__ read_vectors, const int* __restrict__ read_positions,
             float* __restrict__ iface_ws, float* __restrict__ q_ws,
             float* __restrict__ rv_ws, float* __restrict__ drv_ws,
             int* __restrict__ ovorder)
{
  const int tid  = threadIdx.x;
  const int wave = tid >> 5;
  const int lane = tid & 31;

  // ---- GEMM: iface = xi(32x512) @ W(512x138), f32 WMMA (exact f32 math) ----
  {
    const int mt = wave & 1;       // 0..1
    const int nt = wave >> 1;      // 0..8
    const int kp = lane >> 4;      // half-wave
    const int ln = lane & 15;
    const int m  = mt * 16 + ln;
    const int n  = nt * 16 + ln;
    const int nc = (n < IFACE_D) ? n : (IFACE_D - 1);   // clamp -> branchless loads
    const float* __restrict__ xp = xi + m * IND + 2 * kp;  // {xp[k0], xp[k0+1]}
    const float* __restrict__ wp = W + (size_t)(2 * kp) * IFACE_D + nc;
    v8f acc = {0.f, 0.f, 0.f, 0.f, 0.f, 0.f, 0.f, 0.f};
    for (int k0 = 0; k0 < IND; k0 += 4) {
      v2f a;
      a.x = xp[k0];
      a.y = xp[k0 + 1];
      v2f bb;
      bb.x = wp[(size_t)k0 * IFACE_D];
      bb.y = wp[(size_t)k0 * IFACE_D + IFACE_D];
      acc = __builtin_amdgcn_wmma_f32_16x16x4_f32(false, a, false, bb,
                                                  (short)0, acc, false, false);
    }
#pragma unroll
    for (int j = 0; j < 8; ++j) {
      const int row = mt * 16 + j + 8 * kp;
      const int col = nt * 16 + ln;
      const float bv = bias[(col < IFACE_D) ? col : (IFACE_D - 1)];
      iface_ws[row * IFACE_P + col] = acc[j] + bv;   // padded store; cols>=138 unused
    }
  }
  __syncthreads();   // iface_ws visible block-wide (global + barrier semantics)

  // ---- read_query ----
  for (int idx = tid; idx < Bsz * WcD; idx += 576)
    q_ws[idx] = iface_ws[(idx >> 6) * IFACE_P + (idx & 63)];

  // ---- rv = read_vectors + ww * write_vector ----
  for (int idx = tid; idx < Bsz * RD * WcD; idx += 576) {
    const int b   = idx / (RD * WcD);
    const int rem = idx - b * (RD * WcD);
    const int r   = rem >> 6;
    const int w   = rem & 63;
    const float ig   = iface_ws[b * IFACE_P + 2 * WcD + r];
    const float gate = 1.f / (1.f + expf(-iface_ws[b * IFACE_P + (IFACE_D - 1)]));
    const float rw   = read_weights[b * RD + r];
    const float ww   = gate * (ig * rw + (1.f - ig));
    rv_ws[idx] = read_vectors[idx] + ww * iface_ws[b * IFACE_P + WcD + w];
  }

  // ---- scatter winner resolution: order[p] = max flat index writing p ----
  if (tid < Bsz * RD) {
    atomicMax(&ovorder[read_positions[tid]], tid);   // winner r = ovorder % 9
  }

  // ---- d_rv[b][r] = ||rv - q||^2 via same expansion as the reference ----
  if (tid < Bsz * RD) {
    const int b = tid / RD;
    const int r = tid - b * RD;
    const float ig   = iface_ws[b * IFACE_P + 2 * WcD + r];
    const float gate = 1.f / (1.f + expf(-iface_ws[b * IFACE_P + (IFACE_D - 1)]));
    const float rw   = read_weights[b * RD + r];
    const float ww   = gate * (ig * rw + (1.f - ig));
    float ss = 0.f, sq = 0.f, qn = 0.f;
    for (int w = 0; w < WcD; ++w) {
      const float rvv = read_vectors[tid * WcD + w] + ww * iface_ws[b * IFACE_P + WcD + w];
      const float qv  = iface_ws[b * IFACE_P + w];
      ss += rvv * rvv; sq += rvv * qv; qn += qv * qv;
    }
    drv_ws[tid] = ss - 2.f * sq + qn;
  }
}

// ---------------------------------------------------------------------------
// Kernel 2: stream 512 MB of `sparse` once (NT 128-bit loads), per-lane full
// row -> distance -> register top-8; wave32 xor-butterfly + LDS reduction.
// Grid: (32 blocks-per-batch, 32 batches) x 256 threads = 8 rows/thread.
// HBM roofline: 512 MB / 23.3 TB/s ~= 22 us; everything else is noise.
// ---------------------------------------------------------------------------
__global__ __launch_bounds__(256)
void sm_dist_topk(const float* __restrict__ sparse, const float* __restrict__ q_ws,
                  const float* __restrict__ drv_ws, const int* __restrict__ ovorder,
                  float* __restrict__ candd, int* __restrict__ candi)
{
  const int b   = blockIdx.y;
  const int blk = blockIdx.x;
  const int t   = threadIdx.x;

  __shared__ float sdrv[RD];
  if (t < RD) sdrv[t] = drv_ws[b * RD + t];
  __syncthreads();

  // query held in 64 VGPRs (broadcast loads, L2 hits)
  v4f q[16];
  const v4f* q4 = (const v4f*)(q_ws + b * WcD);
#pragma unroll
  for (int c = 0; c < 16; ++c) q[c] = q4[c];
  float qn = 0.f;
#pragma unroll
  for (int c = 0; c < 16; ++c)
    qn += q[c].x * q[c].x + q[c].y * q[c].y + q[c].z * q[c].z + q[c].w * q[c].w;

  float bd[8]; int bi[8];
#pragma unroll
  for (int j = 0; j < 8; ++j) { bd[j] = __builtin_huge_valf(); bi[j] = 0x7fffffff; }

  const float* sb = sparse + (size_t)b * ((size_t)Mmem * WcD);
  const int m0 = blk * 2048 + t;
  for (int i = 0; i < 8; ++i) {
    const int m = m0 + i * 256;
    const v4f* rp = (const v4f*)(sb + (size_t)m * WcD);
    if (i < 7) __builtin_prefetch((const void*)(sb + (size_t)(m + 256) * WcD), 0, 0);
    float ss = 0.f, sq = 0.f;
#pragma unroll
    for (int c = 0; c < 16; ++c) {
      const v4f s = __builtin_nontemporal_load(rp + c);   // stream, don't cache far
      ss += s.x * s.x + s.y * s.y + s.z * s.z + s.w * s.w;
      sq += s.x * q[c].x + s.y * q[c].y + s.z * q[c].z + s.w * q[c].w;
    }
    float d = ss - 2.f * sq + qn;                         // reference's expansion
    const int ov = ovorder[m];
    if (ov >= 0) d = sdrv[ov % RD];                       // overridden row
    cinsert(bd, bi, d, m);
  }

  // wave32 xor-butterfly: disjoint origin groups at every step -> no dupes
#pragma unroll
  for (int off = 16; off >= 1; off >>= 1) {
    float od[8]; int oi[8];
#pragma unroll
    for (int j = 0; j < 8; ++j) {
      od[j] = __shfl_xor(bd[j], off, 32);
      oi[j] = __shfl_xor(bi[j], off, 32);
    }
#pragma unroll
    for (int j = 0; j < 8; ++j) cinsert(bd, bi, od[j], oi[j]);
  }

  __shared__ float swd[8][8];
  __shared__ int   swi[8][8];
  const int wv = t >> 5;
  if ((t & 31) == 0) {
#pragma unroll
    for (int j = 0; j < 8; ++j) { swd[wv][j] = bd[j]; swi[wv][j] = bi[j]; }
  }
  __syncthreads();
  if (t == 0) {
    for (int w2 = 1; w2 < 8; ++w2) {
#pragma unroll
      for (int j = 0; j < 8; ++j) cinsert(bd, bi, swd[w2][j], swi[w2][j]);
    }
    const int cbase = (b * 32 + blk) * 8;
#pragma unroll
    for (int j = 0; j < 8; ++j) { candd[cbase + j] = bd[j]; candi[cbase + j] = bi[j]; }
  }
}

// ---------------------------------------------------------------------------
// Kernel 3: per-batch final merge (256 candidates), dist normalization,
// 9-row gather (rv for overridden positions), write all three outputs.
// ---------------------------------------------------------------------------
__global__ __launch_bounds__(64)
void sm_finalize(const float* __restrict__ sparse, const float* __restrict__ candd,
                 const int* __restrict__ candi, const int* __restrict__ ovorder,
                 const float* __restrict__ rv_ws, const int* __restrict__ last_used,
                 float* __restrict__ out)
{
  const int b = blockIdx.x;
  const int t = threadIdx.x;
  __shared__ int   spos[RD];
  __shared__ float sdist[RD];
  if (t == 0) {
    float bd[8]; int bi[8];
#pragma unroll
    for (int j = 0; j < 8; ++j) { bd[j] = __builtin_huge_valf(); bi[j] = 0x7fffffff; }
    const int base = b * 256;
    for (int j = 0; j < 256; ++j) cinsert(bd, bi, candd[base + j], candi[base + j]);
    const float mx = fmaxf(bd[7], 0.0f);   // max over 8 dists and appended 0
#pragma unroll
    for (int j = 0; j < 8; ++j) { spos[j] = bi[j]; sdist[j] = bd[j] / mx; }
    spos[8]  = last_used[b];
    sdist[8] = 0.0f / mx;                  // reproduces reference 0/max exactly
  }
  __syncthreads();

  float* __restrict__ out1 = out;                          // (32,8,64)
  float* __restrict__ out2 = out + Bsz * KD * WcD;         // (32,1,9)
  float* __restrict__ out3 = out2 + Bsz * RD;              // (32,9,64)
  if (t < RD) out2[b * RD + t] = sdist[t];
  for (int j = 0; j < RD; ++j) {
    const int row = spos[j];
    const int ov  = ovorder[row];
    float v;
    if (ov >= 0) v = rv_ws[(size_t)(b * RD + (ov % RD)) * WcD + t];
    else         v = sparse[((size_t)b * Mmem + row) * WcD + t];
    out3[(b * RD + j) * WcD + t] = v;
    if (j < KD) out1[(b * KD + j) * WcD + t] = v;
  }
}

// ---------------------------------------------------------------------------
extern "C" void kernel_launch(void* const* d_in, const int* in_sizes, int n_in,
                              void* d_out, int out_size, void* d_ws, size_t ws_size,
                              hipStream_t stream) {
  (void)in_sizes; (void)n_in; (void)out_size; (void)ws_size;
  const float* xi             = (const float*)d_in[0];
  const float* sparse         = (const float*)d_in[1];
  const float* read_weights   = (const float*)d_in[2];
  const float* read_vectors   = (const float*)d_in[3];
  const float* W              = (const float*)d_in[4];
  const float* bias           = (const float*)d_in[5];
  const int*   read_positions = (const int*)d_in[6];
  const int*   last_used      = (const int*)d_in[7];
  float* out = (float*)d_out;

  char* ws = (char*)d_ws;                       // ~420 KB total
  float* iface_ws = (float*)(ws + 0);           // 32*144 f32   = 18432 B
  float* q_ws     = (float*)(ws + 18432);       // 32*64  f32   =  8192 B
  float* rv_ws    = (float*)(ws + 26624);       // 32*9*64 f32  = 73728 B
  float* drv_ws   = (float*)(ws + 100352);      // 32*9  f32    =  1152 B
  float* candd    = (float*)(ws + 101504);      // 32*32*8 f32  = 32768 B
  int*   candi    = (int*)  (ws + 134272);      // 32*32*8 i32  = 32768 B
  int*   ovorder  = (int*)  (ws + 167040);      // 65536 i32    = 262144 B

  hipMemsetAsync(ovorder, 0xFF, (size_t)Mmem * sizeof(int), stream);   // all -1
  hipLaunchKernelGGL(sm_prep, dim3(1), dim3(576), 0, stream,
                     xi, W, bias, read_weights, read_vectors, read_positions,
                     iface_ws, q_ws, rv_ws, drv_ws, ovorder);
  hipLaunchKernelGGL(sm_dist_topk, dim3(32, Bsz), dim3(256), 0, stream,
                     sparse, q_ws, drv_ws, ovorder, candd, candi);
  hipLaunchKernelGGL(sm_finalize, dim3(Bsz), dim3(64), 0, stream,
                     sparse, candd, candi, ovorder, rv_ws, last_used, out);
}